// _GraphConv_24172075942104
// MI455X (gfx1250) — compile-verified
//
#include <hip/hip_runtime.h>

#define NN 16
#define JN 17
#define CN 16
#define HN 64
#define WN 64
#define HWN (HN * WN)      // 4096
#define JC (JN * CN)       // 272
#define KK (CN * 9)        // 144  (c_in * 3 * 3)
#define EPSV 1e-5f

#define TILE_H 8           // output rows per block
#define TROWS (TILE_H + 2) // 10 rows incl. halo
#define TCOLS (WN + 2)     // 66 cols incl. halo
#define TSTRIDE (TROWS * TCOLS)  // 660 floats per channel

typedef __attribute__((ext_vector_type(2))) float v2f;
typedef __attribute__((ext_vector_type(8))) float v8f;

// LDS word offset of input element for reduction index kk = ci*9 + kh*3 + kw
// (used only with kk in [0,72); lane-half 1 adds 8*TSTRIDE = +8 channels)
constexpr int xoff(int kk) {
    return (kk / 9) * TSTRIDE + ((kk % 9) / 3) * TCOLS + ((kk % 9) % 3);
}

// ---------------------------------------------------------------------------
// Kernel 0: row-softmax of adjacency [17 x 17]
// ---------------------------------------------------------------------------
__global__ void k_softmax_adj(const float* __restrict__ adj,
                              float* __restrict__ asoft) {
    int r = threadIdx.x;
    if (r < JN) {
        float m = -1e30f;
        for (int c = 0; c < JN; ++c) m = fmaxf(m, adj[r * JN + c]);
        float e[JN];
        float s = 0.f;
        for (int c = 0; c < JN; ++c) { e[c] = __expf(adj[r * JN + c] - m); s += e[c]; }
        float inv = 1.f / s;
        for (int c = 0; c < JN; ++c) asoft[r * JN + c] = e[c] * inv;
    }
}

// ---------------------------------------------------------------------------
// Kernel 1: grouped 3x3 conv via chained V_WMMA_F32_16X16X4_F32
// Block = one (n, j) and 8 rows; LDS-staged halo tile; wave = one row,
// 4 segments of 16 pixels; K = 144 fully unrolled -> 36 static WMMAs.
// K is permuted so lane-half 0 reduces c_in 0..7 and lane-half 1 reduces
// c_in 8..15: all DS offsets become compile-time immediates.
// Weights staged with global_load_async_to_lds_b32 (ASYNCcnt path).
// ---------------------------------------------------------------------------
__global__ void __launch_bounds__(256)
k_gconv(const float* __restrict__ x, const float* __restrict__ Wg,
        const float* __restrict__ bg, float* __restrict__ Bout) {
    __shared__ float sW[CN * KK];          // 16 x 144 weights   (9216 B)
    __shared__ float sX[CN * TSTRIDE];     // 16 ch x 10 x 66    (42240 B)

    const int bid = blockIdx.x;
    const int hb  = bid & 7;               // 8 row-tiles of 8 rows
    const int nj  = bid >> 3;
    const int n   = nj / JN;
    const int j   = nj % JN;
    const int h0  = hb * TILE_H;

    const float* Wj = Wg + (size_t)(j * CN) * KK;
    const float* xj = x + (size_t)(n * JN + j) * CN * HWN;

    // ---- async-stage this joint's weights into LDS (always in bounds) ----
    {
        const unsigned lds0 = (unsigned)(size_t)&sW[0];
        for (int k = threadIdx.x; k < CN * KK; k += 256) {
            unsigned ldsa = lds0 + 4u * (unsigned)k;
            unsigned goff = 4u * (unsigned)k;
            asm volatile("global_load_async_to_lds_b32 %0, %1, %2"
                         :: "v"(ldsa), "v"(goff), "s"(Wj) : "memory");
        }
    }

    // ---- zero the always-out-of-image column halo (ww = 0 and 65) ----
    for (int t = threadIdx.x; t < CN * TROWS * 2; t += 256) {
        int ci = t / (TROWS * 2);
        int r  = t - ci * (TROWS * 2);
        int hh = r >> 1;
        int ww = (r & 1) * (TCOLS - 1);
        sX[ci * TSTRIDE + hh * TCOLS + ww] = 0.f;
    }

    // ---- stage interior + row halo with aligned float4 (b128) loads ----
    // chunks: 16 ch x 10 rows x 16 four-wide column groups = 2560
    for (int t = threadIdx.x; t < CN * TROWS * (WN / 4); t += 256) {
        int ci = t / (TROWS * (WN / 4));
        int r  = t - ci * (TROWS * (WN / 4));
        int hh = r >> 4;
        int c4 = (r & 15) << 2;
        int gh = h0 + hh - 1;
        float4 v = make_float4(0.f, 0.f, 0.f, 0.f);
        if ((unsigned)gh < (unsigned)HN)
            v = *(const float4*)(xj + (size_t)ci * HWN + gh * WN + c4);
        float* d = &sX[ci * TSTRIDE + hh * TCOLS + 1 + c4];
        d[0] = v.x; d[1] = v.y; d[2] = v.z; d[3] = v.w;
    }
    asm volatile("s_wait_asynccnt 0x0" ::: "memory");
    __syncthreads();

    const int wave = threadIdx.x >> 5;
    const int lane = threadIdx.x & 31;
    const int half = lane >> 4;            // which K-pair this lane holds
    const int lm   = lane & 15;            // M row (A) / pixel col (B)
    const int hrow = wave;                 // output row within tile

    // bias for the 8 output channels this lane-half accumulates
    float bias[8];
    #pragma unroll
    for (int v = 0; v < 8; ++v) bias[v] = bg[j * CN + v + 8 * half];

    // A fragment base: element e = 2t+s at sW[lm*144 + 72*half + e]
    const float* aw = &sW[lm * KK + 72 * half];

    for (int seg = 0; seg < 4; ++seg) {
        const int w0 = seg * 16;
        // B fragment base: half 1 reduces input channels 8..15 (+8*TSTRIDE)
        const float* bx = &sX[hrow * TCOLS + w0 + lm + half * 8 * TSTRIDE];

        v8f acc = {};
        #pragma unroll
        for (int t = 0; t < 36; ++t) {
            v2f a, b;
            a.x = aw[2 * t];
            a.y = aw[2 * t + 1];
            b.x = bx[xoff(2 * t)];        // compile-time DS immediates
            b.y = bx[xoff(2 * t + 1)];
            acc = __builtin_amdgcn_wmma_f32_16x16x4_f32(
                false, a, false, b, (short)0, acc, false, false);
        }

        float* op = Bout + ((size_t)(n * JN + j) * CN + 8 * half) * HWN
                         + (h0 + hrow) * WN + w0 + lm;
        #pragma unroll
        for (int v = 0; v < 8; ++v)
            op[(size_t)v * HWN] = acc[v] + bias[v];
    }
}

// ---------------------------------------------------------------------------
// Kernel 2: attention scores + Z = Sw(17x17) x B(17x16), one wave per pixel.
// Rows 0..15 via five fp32 WMMAs (K padded 17->20), row 16 via VALU.
// ---------------------------------------------------------------------------
__global__ void __launch_bounds__(256)
k_att_gemm(const float* __restrict__ Bbuf, const float* __restrict__ asoft,
           const float* __restrict__ w_att, const float* __restrict__ b_att,
           float* __restrict__ Zbuf) {
    __shared__ float sAdj[JN * JN];
    __shared__ float sAI[8][20];
    __shared__ float sAJ[8][20];
    __shared__ float sWatt[2 * CN];

    for (int i = threadIdx.x; i < JN * JN; i += 256) sAdj[i] = asoft[i];
    if (threadIdx.x < 2 * CN) sWatt[threadIdx.x] = w_att[threadIdx.x];
    __syncthreads();

    const int wave = threadIdx.x >> 5;
    const int lane = threadIdx.x & 31;
    const int half = lane >> 4;
    const int lm   = lane & 15;

    const int gpix = blockIdx.x * 8 + wave;
    const int n    = gpix >> 12;                // / 4096
    const int hw   = gpix & 4095;

    const float* Bp = Bbuf + (size_t)n * JC * HWN + hw;   // (j*C+c)*HW stride

    // lane l < 17 computes a_i[l], a_j[l] (16-long dot products)
    if (lane < 20) {
        float ai = 0.f, aj = 0.f;
        if (lane < JN) {
            for (int c = 0; c < CN; ++c) {
                float bv = Bp[(size_t)(lane * CN + c) * HWN];
                ai = fmaf(bv, sWatt[c], ai);
                aj = fmaf(bv, sWatt[CN + c], aj);
            }
        }
        sAI[wave][lane] = ai;
        sAJ[wave][lane] = aj;
    }
    __syncthreads();

    const float batt = b_att[0];
    const float aiI  = sAI[wave][lm];           // row i = lm

    v8f acc = {};
    #pragma unroll
    for (int t = 0; t < 5; ++t) {
        int j0 = 4 * t + 2 * half;
        v2f a, b;
        #pragma unroll
        for (int s = 0; s < 2; ++s) {
            int jj = j0 + s;
            float sw = 0.f, bv = 0.f;
            if (jj < JN) {
                float e = aiI + sAJ[wave][jj] + batt;
                sw = (1.f / (1.f + __expf(-e))) * sAdj[lm * JN + jj];
                bv = Bp[(size_t)(jj * CN + lm) * HWN];
            }
            if (s == 0) { a.x = sw; b.x = bv; } else { a.y = sw; b.y = bv; }
        }
        acc = __builtin_amdgcn_wmma_f32_16x16x4_f32(
            false, a, false, b, (short)0, acc, false, false);
    }

    // row i = 16 (lanes 0..15, c = lm) on the VALU
    float z16 = 0.f;
    if (half == 0) {
        float ai16 = sAI[wave][16];
        for (int jj = 0; jj < JN; ++jj) {
            float e  = ai16 + sAJ[wave][jj] + batt;
            float sw = (1.f / (1.f + __expf(-e))) * sAdj[16 * JN + jj];
            z16 = fmaf(sw, Bp[(size_t)(jj * CN + lm) * HWN], z16);
        }
    }

    float* Zp = Zbuf + (size_t)n * JC * HWN + hw;
    #pragma unroll
    for (int v = 0; v < 8; ++v) {
        int i = v + 8 * half;
        Zp[(size_t)(i * CN + lm) * HWN] = acc[v];
    }
    if (half == 0) Zp[(size_t)(16 * CN + lm) * HWN] = z16;
}

// ---------------------------------------------------------------------------
// Kernel 3: per-channel batch mean / rsqrt(var+eps), float4 streaming
// ---------------------------------------------------------------------------
__global__ void __launch_bounds__(256)
k_bnstats(const float* __restrict__ Zbuf, float* __restrict__ meanb,
          float* __restrict__ rstdb) {
    __shared__ float s1[256];
    __shared__ float s2[256];
    const int ch = blockIdx.x;                  // 0..271
    float sum = 0.f, sq = 0.f;
    for (int idx = threadIdx.x; idx < NN * HWN / 4; idx += 256) {
        int n   = idx >> 10;                    // / (HWN/4)
        int hw4 = (idx & 1023) << 2;
        const float4 v = *(const float4*)&Zbuf[((size_t)n * JC + ch) * HWN + hw4];
        sum += v.x + v.y + v.z + v.w;
        sq   = fmaf(v.x, v.x, fmaf(v.y, v.y, fmaf(v.z, v.z, fmaf(v.w, v.w, sq))));
    }
    s1[threadIdx.x] = sum;
    s2[threadIdx.x] = sq;
    __syncthreads();
    for (int off = 128; off > 0; off >>= 1) {
        if (threadIdx.x < off) {
            s1[threadIdx.x] += s1[threadIdx.x + off];
            s2[threadIdx.x] += s2[threadIdx.x + off];
        }
        __syncthreads();
    }
    if (threadIdx.x == 0) {
        const float inv = 1.f / (float)(NN * HWN);
        float m   = s1[0] * inv;
        float var = s2[0] * inv - m * m;
        meanb[ch] = m;
        rstdb[ch] = rsqrtf(var + EPSV);
    }
}

// ---------------------------------------------------------------------------
// Kernel 4: normalize * gamma + beta, ReLU -> d_out (float4 streaming)
// ---------------------------------------------------------------------------
__global__ void __launch_bounds__(256)
k_bnrelu(const float* __restrict__ Zbuf, const float* __restrict__ meanb,
         const float* __restrict__ rstdb, const float* __restrict__ gamma,
         const float* __restrict__ beta, float* __restrict__ out) {
    size_t i4 = ((size_t)blockIdx.x * 256 + threadIdx.x) * 4;
    int ch = (int)((i4 >> 12) % JC);            // HWN = 4096 divides 4-groups
    float m = meanb[ch], rs = rstdb[ch], g = gamma[ch], b = beta[ch];
    float4 v = *(const float4*)&Zbuf[i4];
    float4 r;
    r.x = fmaxf((v.x - m) * rs * g + b, 0.f);
    r.y = fmaxf((v.y - m) * rs * g + b, 0.f);
    r.z = fmaxf((v.z - m) * rs * g + b, 0.f);
    r.w = fmaxf((v.w - m) * rs * g + b, 0.f);
    *(float4*)&out[i4] = r;
}

// ---------------------------------------------------------------------------
extern "C" void kernel_launch(void* const* d_in, const int* in_sizes, int n_in,
                              void* d_out, int out_size, void* d_ws, size_t ws_size,
                              hipStream_t stream) {
    const float* x     = (const float*)d_in[0];
    const float* adj   = (const float*)d_in[1];
    const float* Wg    = (const float*)d_in[2];
    const float* bg    = (const float*)d_in[3];
    const float* w_att = (const float*)d_in[4];
    const float* b_att = (const float*)d_in[5];
    const float* gamma = (const float*)d_in[6];
    const float* beta  = (const float*)d_in[7];
    float* out = (float*)d_out;

    const size_t elems = (size_t)NN * JC * HWN;   // 17,825,792
    float* ws    = (float*)d_ws;
    float* Bbuf  = ws;                 // conv output B4
    float* Zbuf  = Bbuf + elems;       // attention-aggregated Z
    float* asoft = Zbuf + elems;       // softmax(adj) [17x17]
    float* meanb = asoft + JN * JN;    // [272]
    float* rstdb = meanb + JC;         // [272]

    hipLaunchKernelGGL(k_softmax_adj, dim3(1), dim3(32), 0, stream, adj, asoft);
    hipLaunchKernelGGL(k_gconv, dim3(NN * JN * 8), dim3(256), 0, stream,
                       x, Wg, bg, Bbuf);
    hipLaunchKernelGGL(k_att_gemm, dim3(NN * HWN / 8), dim3(256), 0, stream,
                       Bbuf, asoft, w_att, b_att, Zbuf);
    hipLaunchKernelGGL(k_bnstats, dim3(JC), dim3(256), 0, stream,
                       Zbuf, meanb, rstdb);
    hipLaunchKernelGGL(k_bnrelu, dim3((unsigned)(elems / 1024)), dim3(256), 0,
                       stream, Zbuf, meanb, rstdb, gamma, beta, out);
}